// BorderWeightedLoss_68564857913402
// MI455X (gfx1250) — compile-verified
//
#include <hip/hip_runtime.h>

#define Bk 8
#define Ck 4
#define Hk 512
#define Wk 512
#define HWk (Hk*Wk)           // 1<<18
#define NPIX (Bk*HWk)         // 2097152
#define NPLANE (Bk*Ck)        // 32
#define NBLK_PIX (NPIX/256)   // 8192
#define COLS 16
#define MROWS 16
#define DBIGF 1.0e6f
#define BIG2F 1.0e12f

typedef unsigned int v4u __attribute__((ext_vector_type(4)));
typedef int          v4i __attribute__((ext_vector_type(4)));
typedef int          v8i __attribute__((ext_vector_type(8)));
typedef float        v2f __attribute__((ext_vector_type(2)));
typedef float        v8f __attribute__((ext_vector_type(8)));

// ---------------- Kernel 1: per-pixel CE, label, per-(b,c) pixel counts ----------------
__global__ __launch_bounds__(256) void k_ce(const float* __restrict__ pred,
                                            const float* __restrict__ tgt,
                                            float* __restrict__ ce,
                                            unsigned char* __restrict__ label,
                                            int* __restrict__ cnts) {
    int p = blockIdx.x * 256 + threadIdx.x;
    int b = p >> 18;
    int pix = p & (HWk - 1);
    size_t base = ((size_t)b * Ck) << 18;

    float t0 = tgt[base + pix];
    float t1 = tgt[base + HWk + pix];
    float t2 = tgt[base + 2 * HWk + pix];
    float t3 = tgt[base + 3 * HWk + pix];
    int lbl = 0; float bestv = t0;
    if (t1 > bestv) { bestv = t1; lbl = 1; }
    if (t2 > bestv) { bestv = t2; lbl = 2; }
    if (t3 > bestv) { bestv = t3; lbl = 3; }

    float p0 = pred[base + pix];
    float p1 = pred[base + HWk + pix];
    float p2 = pred[base + 2 * HWk + pix];
    float p3 = pred[base + 3 * HWk + pix];
    float mx = fmaxf(fmaxf(p0, p1), fmaxf(p2, p3));
    float se = expf(p0 - mx) + expf(p1 - mx) + expf(p2 - mx) + expf(p3 - mx);
    float lse = mx + logf(se);
    float pl = (lbl == 0) ? p0 : (lbl == 1) ? p1 : (lbl == 2) ? p2 : p3;

    ce[p] = lse - pl;
    label[p] = (unsigned char)lbl;

    __shared__ int hc[4];
    if (threadIdx.x < 4) hc[threadIdx.x] = 0;
    __syncthreads();
    atomicAdd(&hc[lbl], 1);
    __syncthreads();
    if (threadIdx.x < 4) atomicAdd(&cnts[b * Ck + (int)threadIdx.x], hc[threadIdx.x]);
}

// ---------------- Kernel 2: exact 1D row EDT (matches reference cummax formulation) ----
__global__ __launch_bounds__(256) void k_rows(const unsigned char* __restrict__ label,
                                              float* __restrict__ g2) {
    int t = blockIdx.x * 256 + threadIdx.x;   // 0..16383
    int plane = t >> 9, row = t & 511;
    int b = plane >> 2; unsigned c = (unsigned)(plane & 3);
    const unsigned* lw = (const unsigned*)(label + ((size_t)b << 18) + ((size_t)row << 9));
    float* out = g2 + ((size_t)plane << 18) + ((size_t)row << 9);

    // forward: dl = col - cummax(mask ? col : -1e6)
    float last = -DBIGF;
    for (int w = 0; w < 128; ++w) {
        unsigned u = lw[w];
#pragma unroll
        for (int bi = 0; bi < 4; ++bi) {
            int col = w * 4 + bi;
            float cf = (float)col;
            if (((u >> (8 * bi)) & 255u) == c) last = cf;
            out[col] = cf - last;
        }
    }
    // backward: dr = -cummax_rev(mask ? -col : -1e6) - col ; g2 = min(dl,dr,1e6)^2
    float m = -DBIGF;
    for (int w = 127; w >= 0; --w) {
        unsigned u = lw[w];
#pragma unroll
        for (int bi = 3; bi >= 0; --bi) {
            int col = w * 4 + bi;
            float cf = (float)col;
            if (((u >> (8 * bi)) & 255u) == c) m = -cf;  // -cf > previous m always
            float dr = -m - cf;
            float g = fminf(fminf(out[col], dr), DBIGF);
            out[col] = g * g;
        }
    }
}

// ---------------- Kernel 3: min-plus envelope along columns, TDM-fed LDS tiles --------
// D2[i,j] = min_k (g2[k,j] + k^2) - 2*i*k   [+ i^2 added at the end]
__global__ __launch_bounds__(256) void k_cols(const float* __restrict__ g2,
                                              float* __restrict__ D2) {
    __shared__ float tile[Hk * COLS];   // 32KB: [k][j], j = 16 contiguous columns
    int bx = blockIdx.x;
    int ih = bx & 1;           // row half (0..1)
    int ct = (bx >> 1) & 31;   // column tile (0..31)
    int plane = bx >> 6;       // 0..31

    const float* src = g2 + (((size_t)plane) << 18) + (size_t)ct * COLS;
    unsigned long long ga = (unsigned long long)(const void*)src;
    unsigned ldsoff = (unsigned)(unsigned long long)(void*)&tile[0];

    // Tensor DMA descriptor: 2D tile, 4-byte elems, tile 16x512, tensor 512x512, stride 512
    v4u g0;
    g0[0] = 1u;                                              // count=1, user mode
    g0[1] = ldsoff;                                          // lds_addr
    g0[2] = (unsigned)ga;                                    // global_addr[31:0]
    g0[3] = ((unsigned)(ga >> 32) & 0x01FFFFFFu) | (2u << 30); // global_addr[56:32] | type=2
    v8i g1;
    g1[0] = (int)(2u << 16);                                 // data_size=2 (4B)
    g1[1] = (int)(((unsigned)Wk & 0xFFFFu) << 16);           // tensor_dim0[15:0]=512
    g1[2] = (int)((((unsigned)Wk >> 16) & 0xFFFFu) | (((unsigned)Hk & 0xFFFFu) << 16)); // dim0 hi | dim1 lo
    g1[3] = (int)((((unsigned)Hk >> 16) & 0xFFFFu) | ((unsigned)COLS << 16));           // dim1 hi | tile_dim0=16
    g1[4] = (int)(unsigned)Hk;                               // tile_dim1=512, tile_dim2=0
    g1[5] = (int)(unsigned)Wk;                               // tensor_dim0_stride[31:0]=512
    g1[6] = 0;                                               // stride hi | dim1_stride lo
    g1[7] = 0;
    v4i gz = {0, 0, 0, 0};
    v8i gz8 = {0, 0, 0, 0, 0, 0, 0, 0};

    if (__builtin_amdgcn_readfirstlane((int)(threadIdx.x >> 5)) == 0) {
        __builtin_amdgcn_tensor_load_to_lds(g0, g1, gz, gz, gz8, 0);
        __builtin_amdgcn_s_wait_tensorcnt(0);
    }
    __syncthreads();

    // h[k][j] = g2 + k^2 (exact fp32 integers in the finite regime)
    for (int idx = threadIdx.x; idx < Hk * COLS; idx += 256) {
        float kf = (float)(idx >> 4);
        tile[idx] += kf * kf;
    }
    __syncthreads();

    int j = threadIdx.x & (COLS - 1);
    int ig = threadIdx.x >> 4;                // 0..15
    int i0 = ih * 256 + ig * MROWS;

    float acc[MROWS], coef[MROWS];
#pragma unroll
    for (int m = 0; m < MROWS; ++m) { acc[m] = BIG2F; coef[m] = -2.0f * (float)(i0 + m); }

    float kf = 0.0f;
    for (int k = 0; k < Hk; ++k) {
        float h = tile[k * COLS + j];
#pragma unroll
        for (int m = 0; m < MROWS; ++m)
            acc[m] = fminf(acc[m], fmaf(coef[m], kf, h));   // 1 fma + 1 min per candidate
        kf += 1.0f;
    }

    int jg = ct * COLS + j;
    float* dst = D2 + (((size_t)plane) << 18) + jg;
#pragma unroll
    for (int m = 0; m < MROWS; ++m) {
        float im = (float)(i0 + m);
        dst[(size_t)(i0 + m) * Wk] = fmaf(im, im, acc[m]);
    }
}

// ---------------- Kernel 4: border weights + per-block partial sums -------------------
__global__ __launch_bounds__(256) void k_w(const float* __restrict__ D2,
                                           const float* __restrict__ ce,
                                           const int* __restrict__ cnts,
                                           float* __restrict__ partials) {
    int p = blockIdx.x * 256 + threadIdx.x;
    int b = p >> 18;
    int pix = p & (HWk - 1);

    float d[4];
#pragma unroll
    for (int c = 0; c < 4; ++c) {
        int cnt = cnts[b * Ck + c];
        bool valid = (cnt > 0) && (cnt < HWk);
        float dd = sqrtf(D2[(((size_t)b * Ck + c) << 18) + pix]);
        d[c] = valid ? dd : DBIGF;
    }
    float lo01 = fminf(d[0], d[1]), hi01 = fmaxf(d[0], d[1]);
    float lo23 = fminf(d[2], d[3]), hi23 = fmaxf(d[2], d[3]);
    float d1 = fminf(lo01, lo23);
    float d2 = fminf(fmaxf(lo01, lo23), fminf(hi01, hi23));
    float s = d1 + d2;
    float wgt = 1.0f + 10.0f * expf(-(s * s) / 50.0f);
    float val = ce[p] * (1.0f + wgt);   // base + weighted, fused

    __shared__ float red[256];
    red[threadIdx.x] = val;
    __syncthreads();
    for (int off = 128; off > 0; off >>= 1) {
        if (threadIdx.x < (unsigned)off) red[threadIdx.x] += red[threadIdx.x + off];
        __syncthreads();
    }
    if (threadIdx.x == 0) partials[blockIdx.x] = red[0];
}

// ---------------- Kernel 5: final reduction; 32-lane stage on the matrix unit ---------
__global__ __launch_bounds__(256) void k_final(const float* __restrict__ partials,
                                               float* __restrict__ out) {
    float s = 0.0f;
    for (int i = threadIdx.x; i < NBLK_PIX; i += 256) s += partials[i];
    __shared__ float red[256];
    red[threadIdx.x] = s;
    __syncthreads();
    for (int off = 128; off >= 32; off >>= 1) {
        if (threadIdx.x < (unsigned)off) red[threadIdx.x] += red[threadIdx.x + off];
        __syncthreads();
    }
    // 32 partials -> A matrix (16x4 f32 layout: VGPR0 = K0 / K2 columns), B = ones.
    // D[M][N] = s_M + s_{M+16}; sum of D VGPRs gives rows 0-7 (lanes 0-15) / 8-15 (16-31).
    float v = red[threadIdx.x & 31];
    v2f a; a.x = v;    a.y = 0.0f;
    v2f bb; bb.x = 1.0f; bb.y = 1.0f;
    v8f cz = {0.f, 0.f, 0.f, 0.f, 0.f, 0.f, 0.f, 0.f};
    v8f dm = __builtin_amdgcn_wmma_f32_16x16x4_f32(false, a, false, bb, (short)0, cz,
                                                   false, false);
    float t = dm[0] + dm[1] + dm[2] + dm[3] + dm[4] + dm[5] + dm[6] + dm[7];
    __shared__ float tout[32];
    if (threadIdx.x < 32) tout[threadIdx.x] = t;
    __syncthreads();
    if (threadIdx.x == 0)
        out[0] = (tout[0] + tout[16]) * (1.0f / (float)NPIX);
}

// ---------------- Host launcher -------------------------------------------------------
extern "C" void kernel_launch(void* const* d_in, const int* in_sizes, int n_in,
                              void* d_out, int out_size, void* d_ws, size_t ws_size,
                              hipStream_t stream) {
    const float* pred = (const float*)d_in[0];
    const float* tgt  = (const float*)d_in[1];
    char* ws = (char*)d_ws;

    float* g2            = (float*)(ws);                    // 32 MB
    float* D2            = (float*)(ws + 33554432);         // 32 MB
    float* ce            = (float*)(ws + 67108864);         // 8 MB
    unsigned char* label = (unsigned char*)(ws + 75497472); // 2 MB
    int* cnts            = (int*)(ws + 77594624);           // 128 B
    float* partials      = (float*)(ws + 77594752);         // 32 KB
    float* out           = (float*)d_out;

    (void)hipMemsetAsync(cnts, 0, NPLANE * sizeof(int), stream);
    k_ce   <<<NBLK_PIX,          256, 0, stream>>>(pred, tgt, ce, label, cnts);
    k_rows <<<NPLANE * Hk / 256, 256, 0, stream>>>(label, g2);
    k_cols <<<NPLANE * 32 * 2,   256, 0, stream>>>(g2, D2);
    k_w    <<<NBLK_PIX,          256, 0, stream>>>(D2, ce, cnts, partials);
    k_final<<<1,                 256, 0, stream>>>(partials, out);
}